// VDPDropout_39779987095992
// MI455X (gfx1250) — compile-verified
//
#include <hip/hip_runtime.h>
#include <stdint.h>

// float4 as an ext vector so __builtin_nontemporal_* accept it directly.
typedef float f32x4 __attribute__((ext_vector_type(4)));

#define B_DIM   16
#define HW2     256          // 16*16 flattened spatial
#define C_DIM   64
#define C4      16           // C_DIM / 4
#define ROW_F4  (HW2 * C4)   // float4 per (b) mask row: 256*16 = 4096 (64 KB)
#define MU_N    (B_DIM * HW2 * C_DIM)   // 262144 floats
#define I_PER_BLK 8

// ---------------------------------------------------------------------------
// Kernel A: mu_out = (drop_mask >= 0.2) ? mu_in * 1.25 : 0   (262K floats)
// ---------------------------------------------------------------------------
__global__ void vdp_mu_kernel(const f32x4* __restrict__ mu_in,
                              const f32x4* __restrict__ drop_mask,
                              f32x4* __restrict__ mu_out, int n4) {
  int idx = blockIdx.x * blockDim.x + threadIdx.x;
  if (idx >= n4) return;
  f32x4 mu = mu_in[idx];
  f32x4 dm = drop_mask[idx];
  f32x4 o;
  o.x = (dm.x >= 0.2f) ? mu.x * 1.25f : 0.0f;
  o.y = (dm.y >= 0.2f) ? mu.y * 1.25f : 0.0f;
  o.z = (dm.z >= 0.2f) ? mu.z * 1.25f : 0.0f;
  o.w = (dm.w >= 0.2f) ? mu.w * 1.25f : 0.0f;
  mu_out[idx] = o;  // regular (RT) store: kernel B re-reads this from L2
}

// ---------------------------------------------------------------------------
// Kernel B: Sigma_out[b,i,j,c] = Sigma_in[b,i,j,c] * g[b,i,c] * g[b,j,c]
//           g = (mu_out != 0) ? 1.25 : 0   (so g_i*g_j == 1.5625*m_i*m_j)
// Block = (b, 8 consecutive i rows). mu_out[b,:,:] (64 KB) is staged into LDS
// once per block via CDNA5 async global->LDS DMA; each thread's 16 g_j quads
// are then hoisted into registers so the steady-state loop is pure
// NT-load -> 8 muls -> NT-store.
// ---------------------------------------------------------------------------
__global__ __launch_bounds__(256)
void vdp_sigma_kernel(const float* __restrict__ Sigma_in,
                      const float* __restrict__ mu_out,
                      float* __restrict__ Sigma_out) {
  __shared__ __align__(16) float gmask[ROW_F4 * 4];  // 16384 floats = 64 KB

  const int b  = blockIdx.y;
  const int i0 = blockIdx.x * I_PER_BLK;
  const int t  = threadIdx.x;

  // ---- Stage mu_out[b, :, :] into LDS with async global->LDS DMA ----------
  // GVS mode: uniform SGPR base + per-lane 32-bit byte offset; LDS dest
  // address comes from a VGPR (low 32 bits of the AS(3) pointer).
  const char* gbase = (const char*)(mu_out + (size_t)b * (HW2 * C_DIM));
  uint32_t smbase = (uint32_t)(uintptr_t)(void*)gmask;  // LDS offset (low 32b)
#pragma unroll
  for (int k = 0; k < 16; ++k) {
    uint32_t byteoff = (uint32_t)(t + 256 * k) * 16u;   // 4096 b128 chunks
    uint32_t ldsaddr = smbase + byteoff;
    asm volatile("global_load_async_to_lds_b128 %0, %1, %2"
                 :
                 : "v"(ldsaddr), "v"(byteoff), "s"(gbase)
                 : "memory");
  }
  asm volatile("s_wait_asynccnt 0" ::: "memory");
  __syncthreads();

  // ---- Convert staged mu values to multiplicative factors in place --------
  f32x4* sm4 = (f32x4*)gmask;
#pragma unroll
  for (int k = 0; k < 16; ++k) {
    int idx = t + 256 * k;
    f32x4 v = sm4[idx];
    f32x4 g;
    g.x = (v.x != 0.0f) ? 1.25f : 0.0f;
    g.y = (v.y != 0.0f) ? 1.25f : 0.0f;
    g.z = (v.z != 0.0f) ? 1.25f : 0.0f;
    g.w = (v.w != 0.0f) ? 1.25f : 0.0f;
    sm4[idx] = g;
  }
  __syncthreads();

  // ---- Hoist this thread's 16 g_j quads into registers (invariant over i) -
  const int c4 = t & 15;   // which float4 of the 64 channels
  const int jg = t >> 4;   // j = jj*16 + jg, jj in [0,16)
  f32x4 gj[16];
#pragma unroll
  for (int jj = 0; jj < 16; ++jj) {
    gj[jj] = sm4[(jj * 16 + jg) * C4 + c4];
  }

  // ---- Stream Sigma: NT b128 load -> 8 v_mul_f32 -> NT b128 store ---------
  const f32x4* __restrict__ sin4  = (const f32x4*)Sigma_in;
  f32x4* __restrict__       sout4 = (f32x4*)Sigma_out;

  for (int ii = 0; ii < I_PER_BLK; ++ii) {
    const int i = i0 + ii;
    const f32x4 gi = sm4[i * C4 + c4];
    const size_t rowbase = ((size_t)(b * HW2 + i) * HW2) * C4 + c4;
#pragma unroll
    for (int jj = 0; jj < 16; ++jj) {
      const size_t off = rowbase + (size_t)(jj * 16 + jg) * C4;
      f32x4 s = __builtin_nontemporal_load(&sin4[off]);   // th:NT streaming
      f32x4 o = s * gi * gj[jj];
      __builtin_nontemporal_store(o, &sout4[off]);
    }
  }
}

extern "C" void kernel_launch(void* const* d_in, const int* in_sizes, int n_in,
                              void* d_out, int out_size, void* d_ws, size_t ws_size,
                              hipStream_t stream) {
  const float* mu_in     = (const float*)d_in[0];
  const float* Sigma_in  = (const float*)d_in[1];
  const float* drop_mask = (const float*)d_in[2];

  float* mu_out    = (float*)d_out;          // first 262144 floats
  float* Sigma_out = (float*)d_out + MU_N;   // remaining 67,108,864 floats

  // Kernel A: 65536 float4 elements
  const int n4 = MU_N / 4;
  vdp_mu_kernel<<<dim3((n4 + 255) / 256), dim3(256), 0, stream>>>(
      (const f32x4*)mu_in, (const f32x4*)drop_mask, (f32x4*)mu_out, n4);

  // Kernel B: grid = (256/I_PER_BLK i-tiles, 16 batches), 256 threads (8 waves)
  vdp_sigma_kernel<<<dim3(HW2 / I_PER_BLK, B_DIM), dim3(256), 0, stream>>>(
      Sigma_in, mu_out, Sigma_out);
}